// GteLoss_62285615727094
// MI455X (gfx1250) — compile-verified
//
#include <hip/hip_runtime.h>
#include <math.h>

#define NROWS 8192
#define DIM   64
#define INV_T 20.0f

typedef __attribute__((ext_vector_type(2))) float v2f;
typedef __attribute__((ext_vector_type(8))) float v8f;

// ---------------------------------------------------------------------------
// Kernel 0: zero the partition accumulator and the scalar output.
// ---------------------------------------------------------------------------
__global__ __launch_bounds__(256) void GteLoss_init_kernel(float* __restrict__ partition,
                                                           float* __restrict__ out) {
    int i = blockIdx.x * 256 + threadIdx.x;
    if (i < NROWS) partition[i] = 0.0f;
    if (i == 0) out[0] = 0.0f;
}

// ---------------------------------------------------------------------------
// Kernel 1: row-normalize A and P (one wave32 per row; lane holds float2).
// ---------------------------------------------------------------------------
__global__ __launch_bounds__(256) void GteLoss_norm_kernel(const float* __restrict__ A,
                                                           const float* __restrict__ P,
                                                           float* __restrict__ Ahat,
                                                           float* __restrict__ Phat) {
    int wid  = threadIdx.x >> 5;
    int lane = threadIdx.x & 31;
    int row  = blockIdx.x * 8 + wid;          // 0 .. 2*NROWS-1
    const float* src; float* dst; int r;
    if (row < NROWS) { src = A; dst = Ahat; r = row; }
    else             { src = P; dst = Phat; r = row - NROWS; }

    const v2f* s = (const v2f*)(src + (size_t)r * DIM);
    v2f v = s[lane];
    float ss = v.x * v.x + v.y * v.y;
    ss += __shfl_xor(ss, 1, 32);
    ss += __shfl_xor(ss, 2, 32);
    ss += __shfl_xor(ss, 4, 32);
    ss += __shfl_xor(ss, 8, 32);
    ss += __shfl_xor(ss, 16, 32);
    float inv = rsqrtf(fmaxf(ss, 1e-24f));
    v2f o; o.x = v.x * inv; o.y = v.y * inv;
    ((v2f*)(dst + (size_t)r * DIM))[lane] = o;
}

// ---------------------------------------------------------------------------
// Kernel 2: fused tile GEMM (f32 WMMA) + exp + row/col partition sums.
//   blockIdx.z: 0 -> E_ap (row+col sums), 1 -> E_aa (row sums), 2 -> E_pp (row sums)
//   Block = 256 threads = 8 wave32; 64x64 output tile; full K=64 per tile.
// ---------------------------------------------------------------------------
__global__ __launch_bounds__(256) void GteLoss_tile_kernel(const float* __restrict__ Ahat,
                                                           const float* __restrict__ Phat,
                                                           float* __restrict__ partition) {
    __shared__ float As[64][68];     // pad to 68 floats: 16B-aligned rows, no bank conflicts
    __shared__ float Bs[64][68];
    __shared__ float rowacc[64];
    __shared__ float colacc[64];

    const int tid = threadIdx.x;
    const int bx = blockIdx.x, by = blockIdx.y, z = blockIdx.z;

    const float* Ab = (z == 2 ? Phat : Ahat) + (size_t)by * 64 * DIM;
    const float* Bb = (z == 1 ? Ahat : Phat) + (size_t)bx * 64 * DIM;

    if (tid < 64) { rowacc[tid] = 0.0f; colacc[tid] = 0.0f; }

    // Stage 64x64 tiles of A-rows and B-rows into LDS (float4, coalesced).
#pragma unroll
    for (int it = 0; it < 4; ++it) {
        int slot = tid + it * 256;          // 0..1023 float4 slots
        int row  = slot >> 4;
        int c4   = (slot & 15) * 4;
        *(float4*)&As[row][c4] = *(const float4*)(Ab + row * DIM + c4);
        *(float4*)&Bs[row][c4] = *(const float4*)(Bb + row * DIM + c4);
    }
    __syncthreads();

    const int wid  = tid >> 5;
    const int lane = tid & 31;
    const int tm   = (wid & 3) * 16;        // M offset of this wave's 16x32 strip
    const int n0   = (wid >> 2) * 32;       // N offset
    const int l16  = lane & 15;
    const int lh   = lane >> 4;             // half-wave id (K split for f32 WMMA)
    const int m    = tm + l16;

    v8f acc0 = {}; v8f acc1 = {};
    // ISA 7.12.2 32-bit A 16x4 layout: lanes 0-15 hold K={0,1}, lanes 16-31 K={2,3}.
#pragma unroll
    for (int kk = 0; kk < 16; ++kk) {
        int kb = kk * 4 + lh * 2;
        v2f a  = *(const v2f*)&As[m][kb];                 // A[m][kb..kb+1]
        v2f b0 = *(const v2f*)&Bs[n0 + l16][kb];          // B^T rows -> B[k][n]
        v2f b1 = *(const v2f*)&Bs[n0 + 16 + l16][kb];
        acc0 = __builtin_amdgcn_wmma_f32_16x16x4_f32(false, a, false, b0,
                                                     (short)0, acc0, false, false);
        acc1 = __builtin_amdgcn_wmma_f32_16x16x4_f32(false, a, false, b1,
                                                     (short)0, acc1, false, false);
    }

    // acc[r] @ lane: element (M = tm + 8*lh + r, N = n0 (+16) + l16)
    float colp0 = 0.0f, colp1 = 0.0f;
#pragma unroll
    for (int r = 0; r < 8; ++r) {
        float e0 = __expf(acc0[r] * INV_T);
        float e1 = __expf(acc1[r] * INV_T);
        colp0 += e0;
        colp1 += e1;
        float er = e0 + e1;                  // both N-subtiles feed the same row
        er += __shfl_xor(er, 1, 32);
        er += __shfl_xor(er, 2, 32);
        er += __shfl_xor(er, 4, 32);
        er += __shfl_xor(er, 8, 32);
        if (l16 == 0) atomicAdd(&rowacc[tm + lh * 8 + r], er);
    }
    if (z == 0) {
        colp0 += __shfl_xor(colp0, 16, 32);  // combine M halves of the 16x16 subtile
        colp1 += __shfl_xor(colp1, 16, 32);
        if (lh == 0) {
            atomicAdd(&colacc[n0 + l16], colp0);
            atomicAdd(&colacc[n0 + 16 + l16], colp1);
        }
    }
    __syncthreads();

    if (tid < 64) {
        atomicAdd(&partition[(size_t)by * 64 + tid], rowacc[tid]);
        if (z == 0) atomicAdd(&partition[(size_t)bx * 64 + tid], colacc[tid]);
    }
}

// ---------------------------------------------------------------------------
// Kernel 3: diagonal logits + log(partition) + mean reduction into d_out.
// ---------------------------------------------------------------------------
__global__ __launch_bounds__(256) void GteLoss_final_kernel(const float* __restrict__ Ahat,
                                                            const float* __restrict__ Phat,
                                                            const float* __restrict__ partition,
                                                            float* __restrict__ out) {
    __shared__ float red[256];
    int i = blockIdx.x * 256 + threadIdx.x;
    const float4* a = (const float4*)(Ahat + (size_t)i * DIM);
    const float4* p = (const float4*)(Phat + (size_t)i * DIM);
    float dot = 0.0f;
#pragma unroll
    for (int d = 0; d < 16; ++d) {
        float4 av = a[d], pv = p[d];
        dot += av.x * pv.x + av.y * pv.y + av.z * pv.z + av.w * pv.w;
    }
    float selfterm = __expf(INV_T);
    float part = partition[i] - 2.0f * selfterm;
    float contrib = -(dot * INV_T - __logf(part));
    red[threadIdx.x] = contrib;
    __syncthreads();
    for (int s = 128; s > 0; s >>= 1) {
        if (threadIdx.x < s) red[threadIdx.x] += red[threadIdx.x + s];
        __syncthreads();
    }
    if (threadIdx.x == 0) atomicAdd(out, red[0] * (1.0f / (float)NROWS));
}

// ---------------------------------------------------------------------------
extern "C" void kernel_launch(void* const* d_in, const int* in_sizes, int n_in,
                              void* d_out, int out_size, void* d_ws, size_t ws_size,
                              hipStream_t stream) {
    (void)in_sizes; (void)n_in; (void)out_size; (void)ws_size;
    const float* A = (const float*)d_in[0];
    const float* P = (const float*)d_in[1];
    float* out = (float*)d_out;

    float* ws        = (float*)d_ws;
    float* Ahat      = ws;                                  // NROWS*DIM
    float* Phat      = ws + (size_t)NROWS * DIM;            // NROWS*DIM
    float* partition = ws + 2 * (size_t)NROWS * DIM;        // NROWS

    GteLoss_init_kernel<<<(NROWS + 255) / 256, 256, 0, stream>>>(partition, out);
    GteLoss_norm_kernel<<<(2 * NROWS) / 8, 256, 0, stream>>>(A, P, Ahat, Phat);

    dim3 grid(NROWS / 64, NROWS / 64, 3);
    GteLoss_tile_kernel<<<grid, 256, 0, stream>>>(Ahat, Phat, partition);

    GteLoss_final_kernel<<<NROWS / 256, 256, 0, stream>>>(Ahat, Phat, partition, out);
}